// RESCAL_15040975470743
// MI455X (gfx1250) — compile-verified
//
#include <hip/hip_runtime.h>

typedef __attribute__((ext_vector_type(2))) float v2f;
typedef __attribute__((ext_vector_type(4))) float v4f;
typedef __attribute__((ext_vector_type(8))) float v8f;

#define B_  2
#define E_  1024
#define C_  128
#define R_  32
#define RG  8              // r-values handled per wave
#define NGRP (R_ / RG)     // 4 r-groups

// One wave per (16x16 (s,o) tile, r-group). Wave computes the Gram tile for BOTH
// batches with V_WMMA_F32_16X16X4_F32 (K=128 -> 32 steps each), re-stages the
// accumulators through LDS into row-major per-lane float4 chunks, then streams
// out[b,s,r,o] = G[b,s,o] * R[r,s,o] for its 8 r's using b128 NT loads/stores.
__global__ __launch_bounds__(256) void rescal_fused_kernel(
    const float* __restrict__ x,   // (B, E, C)
    const float* __restrict__ Rt,  // (R, E, E)
    float* __restrict__ out)       // (B, E, R, E)
{
    __shared__ float lds[8 * 512];   // 8 waves * (2 batches * 256 floats) = 16 KB

    const int lane = threadIdx.x & 31;
    const int wib  = threadIdx.x >> 5;                     // wave in block, 0..7
    const int wid  = blockIdx.x * 8 + wib;                 // 0..16383
    const int g    = wid & (NGRP - 1);                     // r-group 0..3
    const int tile = wid >> 2;                             // 0..4095
    const int s0   = (tile >> 6) << 4;                     // tile row * 16
    const int o0   = (tile & 63) << 4;                     // tile col * 16

    const int half = lane >> 4;   // 0 or 1
    const int l16  = lane & 15;

    // ---- Phase A: Gram tiles via WMMA (both batches) -----------------------
    // WMMA f32 16x16x4 lane layout:
    //   A (16x4): lane(l16,half) holds X[s0+l16][k0 + 2*half + {0,1}]
    //   B (4x16): lane(l16,half) holds X[o0+l16][k0 + 2*half + {0,1}]
    const float* a0 = x + (size_t)(s0 + l16) * C_ + 2 * half;
    const float* b0 = x + (size_t)(o0 + l16) * C_ + 2 * half;
    const size_t batch_stride = (size_t)E_ * C_;

    v8f acc0 = {0.f, 0.f, 0.f, 0.f, 0.f, 0.f, 0.f, 0.f};  // batch 0
    v8f acc1 = {0.f, 0.f, 0.f, 0.f, 0.f, 0.f, 0.f, 0.f};  // batch 1

#pragma unroll 4
    for (int k0 = 0; k0 < C_; k0 += 4) {
        v2f aA = *(const v2f*)(a0 + k0);
        v2f bA = *(const v2f*)(b0 + k0);
        v2f aB = *(const v2f*)(a0 + batch_stride + k0);
        v2f bB = *(const v2f*)(b0 + batch_stride + k0);
        // 8 args: (neg_a, A, neg_b, B, c_mod, C, reuse_a, reuse_b)
        acc0 = __builtin_amdgcn_wmma_f32_16x16x4_f32(false, aA, false, bA,
                                                     (short)0, acc0, false, false);
        acc1 = __builtin_amdgcn_wmma_f32_16x16x4_f32(false, aB, false, bB,
                                                     (short)0, acc1, false, false);
    }

    // ---- Phase B: re-stage accumulators through LDS (wave-private) ---------
    // D layout: VGPR v, lanes 0-15 -> row m=v, lanes 16-31 -> row m=v+8; col n=l16.
    float* w = &lds[wib * 512];        // [0..255] batch0 tile, [256..511] batch1 tile
#pragma unroll
    for (int v = 0; v < 8; ++v) {
        const int m = v + 8 * half;
        w[m * 16 + l16]       = acc0[v];
        w[256 + m * 16 + l16] = acc1[v];
    }
    // LDS ops within a wave are in-order: safe to read back without a barrier.
    const int rowa = lane >> 2;        // 0..7
    const int c4   = (lane & 3) * 4;   // column chunk 0,4,8,12
    const v4f g0a = *(const v4f*)&w[rowa * 16 + c4];               // b0, rows 0..7
    const v4f g0b = *(const v4f*)&w[(rowa + 8) * 16 + c4];         // b0, rows 8..15
    const v4f g1a = *(const v4f*)&w[256 + rowa * 16 + c4];         // b1, rows 0..7
    const v4f g1b = *(const v4f*)&w[256 + (rowa + 8) * 16 + c4];   // b1, rows 8..15

    // ---- Phase C: stream out = G * R with b128 NT accesses -----------------
    const size_t out_b_stride = (size_t)E_ * R_ * E_;   // stride over b in out
    const int    rbase0       = g * RG;

    const float* rA = Rt + ((size_t)rbase0 * E_ + (s0 + rowa)) * E_ + o0 + c4;
    const float* rB = rA + (size_t)8 * E_;              // rows +8
    float* oA0 = out + (((size_t)(s0 + rowa) * R_ + rbase0) * E_) + o0 + c4;
    float* oB0 = oA0 + (size_t)8 * R_ * E_;             // rows +8 in s
    float* oA1 = oA0 + out_b_stride;
    float* oB1 = oB0 + out_b_stride;

#pragma unroll
    for (int ri = 0; ri < RG; ++ri) {
        const size_t rOff = (size_t)ri * E_ * E_;       // r stride in R
        const size_t oOff = (size_t)ri * E_;            // r stride in out
        v4f ra = __builtin_nontemporal_load((const v4f*)(rA + rOff));
        v4f rb = __builtin_nontemporal_load((const v4f*)(rB + rOff));
        __builtin_nontemporal_store(g0a * ra, (v4f*)(oA0 + oOff));
        __builtin_nontemporal_store(g0b * rb, (v4f*)(oB0 + oOff));
        __builtin_nontemporal_store(g1a * ra, (v4f*)(oA1 + oOff));
        __builtin_nontemporal_store(g1b * rb, (v4f*)(oB1 + oOff));
    }
}

extern "C" void kernel_launch(void* const* d_in, const int* in_sizes, int n_in,
                              void* d_out, int out_size, void* d_ws, size_t ws_size,
                              hipStream_t stream) {
    const float* x  = (const float*)d_in[0];  // (2, 1024, 128)
    const float* Rt = (const float*)d_in[1];  // (32, 1024, 1024)
    float* out = (float*)d_out;               // (2, 1024, 32, 1024)

    // 16384 waves = 64*64 tiles * 4 r-groups; 8 waves (256 threads) per block.
    rescal_fused_kernel<<<2048, 256, 0, stream>>>(x, Rt, out);
}